// Decoder_76536317215012
// MI455X (gfx1250) — compile-verified
//
#include <hip/hip_runtime.h>
#include <math.h>

#define H   1024
#define V   50257
#define SQ  8192

typedef float f32x2 __attribute__((ext_vector_type(2)));
typedef float f32x8 __attribute__((ext_vector_type(8)));

// ---------------------------------------------------------------------------
// WMMA f32 16x16x4 GEMV inner loop.
// One wave computes 16 output rows. A fragment: lane l holds W[mb+(l&15)][k+2*(l>>4) .. +1]
// B fragment broadcasts vector x into all 16 columns: same (k + 2*(l>>4)) addressing.
// NT=true issues non-temporal loads for single-use streaming weights so the
// 192MB L2 keeps enc resident between its two passes.
// ---------------------------------------------------------------------------
template <bool NT>
__device__ __forceinline__ f32x8 wmma_gemv_accum(const float* __restrict__ wrow,
                                                 const float* __restrict__ x,
                                                 int K, int koff, f32x8 c) {
#pragma unroll 8
  for (int k = 0; k < K; k += 4) {
    f32x2 a = NT ? __builtin_nontemporal_load((const f32x2*)(wrow + k + koff))
                 : *(const f32x2*)(wrow + k + koff);
    f32x2 b = *(const f32x2*)(x + k + koff);
    c = __builtin_amdgcn_wmma_f32_16x16x4_f32(false, a, false, b, (short)0, c,
                                              false, false);
  }
  return c;
}

// ---------------------------------------------------------------------------
// gates = w_ih @ x + w_hh @ h0 + b_ih + b_hh   (4096 rows, K=1024 each term)
// 256 waves (32 blocks x 256 threads), one wave per 16 rows.
// ---------------------------------------------------------------------------
__global__ void gates_kernel(const int* __restrict__ word,
                             const float* __restrict__ emb,
                             const float* __restrict__ h0,
                             const float* __restrict__ w_ih,
                             const float* __restrict__ w_hh,
                             const float* __restrict__ b_ih,
                             const float* __restrict__ b_hh,
                             float* __restrict__ gates) {
  int wave = blockIdx.x * (blockDim.x >> 5) + (threadIdx.x >> 5);
  int lane = threadIdx.x & 31;
  int mb   = wave * 16;
  int m    = mb + (lane & 15);
  int koff = (lane >> 4) * 2;
  const float* x = emb + (size_t)word[0] * H;

  f32x8 c = {};
  c = wmma_gemv_accum<true>(w_ih + (size_t)m * H, x,  H, koff, c);
  c = wmma_gemv_accum<true>(w_hh + (size_t)m * H, h0, H, koff, c);

  if ((lane & 15) == 0) {
    int rbase = mb + (lane >> 4) * 8;
#pragma unroll
    for (int r = 0; r < 8; ++r)
      gates[rbase + r] = c[r] + b_ih[rbase + r] + b_hh[rbase + r];
  }
}

// ---------------------------------------------------------------------------
// LSTM pointwise: c = f*c0 + i*g ; h = o*tanh(c)
// ---------------------------------------------------------------------------
__device__ __forceinline__ float sigmoidf(float v) { return 1.0f / (1.0f + expf(-v)); }

__global__ void lstm_kernel(const float* __restrict__ gates,
                            const float* __restrict__ c0,
                            float* __restrict__ out_h,
                            float* __restrict__ out_c,
                            float* __restrict__ comb) {
  int i = blockIdx.x * blockDim.x + threadIdx.x;   // 0..1023
  float ig = sigmoidf(gates[i]);
  float fg = sigmoidf(gates[H + i]);
  float gg = tanhf(gates[2 * H + i]);
  float og = sigmoidf(gates[3 * H + i]);
  float cc = fg * c0[i] + ig * gg;
  float hh = og * tanhf(cc);
  out_c[i] = cc;
  out_h[i] = hh;
  comb[i]  = hh;          // first half of [h ; attn_out]
}

// ---------------------------------------------------------------------------
// scores = enc @ h   (8192 rows, K=1024).  512 waves (64 blocks x 256 thr).
// enc loads stay regular-temporal: re-read by attnout from L2.
// ---------------------------------------------------------------------------
__global__ void scores_kernel(const float* __restrict__ enc,
                              const float* __restrict__ h,
                              float* __restrict__ scores) {
  int wave = blockIdx.x * (blockDim.x >> 5) + (threadIdx.x >> 5);
  int lane = threadIdx.x & 31;
  int mb   = wave * 16;
  int m    = mb + (lane & 15);
  int koff = (lane >> 4) * 2;

  f32x8 c = {};
  c = wmma_gemv_accum<false>(enc + (size_t)m * H, h, H, koff, c);

  if ((lane & 15) == 0) {
    int rbase = mb + (lane >> 4) * 8;
#pragma unroll
    for (int r = 0; r < 8; ++r) scores[rbase + r] = c[r];
  }
}

// ---------------------------------------------------------------------------
// Block reductions (256 threads, wave32)
// ---------------------------------------------------------------------------
__device__ __forceinline__ float block_reduce_max256(float v) {
  __shared__ float s[8];
#pragma unroll
  for (int o = 16; o > 0; o >>= 1) v = fmaxf(v, __shfl_xor(v, o, 32));
  __syncthreads();
  if ((threadIdx.x & 31) == 0) s[threadIdx.x >> 5] = v;
  __syncthreads();
  float r = s[0];
#pragma unroll
  for (int w = 1; w < 8; ++w) r = fmaxf(r, s[w]);
  return r;
}

__device__ __forceinline__ float block_reduce_sum256(float v) {
  __shared__ float s[8];
#pragma unroll
  for (int o = 16; o > 0; o >>= 1) v += __shfl_xor(v, o, 32);
  __syncthreads();
  if ((threadIdx.x & 31) == 0) s[threadIdx.x >> 5] = v;
  __syncthreads();
  float r = s[0];
#pragma unroll
  for (int w = 1; w < 8; ++w) r += s[w];
  return r;
}

// ---------------------------------------------------------------------------
// softmax over 8192 scores -> attn (single block, 256 threads)
// ---------------------------------------------------------------------------
__global__ void softmax_kernel(const float* __restrict__ scores,
                               float* __restrict__ attn) {
  int tid = threadIdx.x;
  float m = -INFINITY;
  for (int s = tid; s < SQ; s += 256) m = fmaxf(m, scores[s]);
  float M = block_reduce_max256(m);
  float acc = 0.0f;
  for (int s = tid; s < SQ; s += 256) acc += expf(scores[s] - M);
  float S = block_reduce_sum256(acc);
  float inv = 1.0f / S;
  for (int s = tid; s < SQ; s += 256) attn[s] = expf(scores[s] - M) * inv;
}

// ---------------------------------------------------------------------------
// attn_out partials: 128 blocks = 4 column tiles x 32 s-chunks of 256
// enc read here should hit L2 (second pass).
// ---------------------------------------------------------------------------
__global__ void attnout_partial_kernel(const float* __restrict__ attn,
                                       const float* __restrict__ enc,
                                       float* __restrict__ patt) {
  __shared__ float sa[256];
  int tid = threadIdx.x;
  int cb  = (blockIdx.x & 3) * 256;
  int sc0 = (blockIdx.x >> 2) * 256;
  sa[tid] = attn[sc0 + tid];
  __syncthreads();
  int col = cb + tid;
  float acc = 0.0f;
#pragma unroll 8
  for (int j = 0; j < 256; ++j) acc += sa[j] * enc[(size_t)(sc0 + j) * H + col];
  patt[(size_t)(blockIdx.x >> 2) * H + col] = acc;
}

__global__ void attnout_reduce_kernel(const float* __restrict__ patt,
                                      float* __restrict__ comb) {
  int col = blockIdx.x * blockDim.x + threadIdx.x;   // 0..1023
  float acc = 0.0f;
#pragma unroll
  for (int j = 0; j < 32; ++j) acc += patt[(size_t)j * H + col];
  comb[H + col] = acc;          // second half of [h ; attn_out]
}

// ---------------------------------------------------------------------------
// logits = W_out @ comb + b_out   (50257 rows, K=2048).  3142 row tiles.
// W_out is 412MB single-use: non-temporal loads.
// ---------------------------------------------------------------------------
__global__ void logits_kernel(const float* __restrict__ Wout,
                              const float* __restrict__ comb,
                              const float* __restrict__ bout,
                              float* __restrict__ logits) {
  const int TILES = (V + 15) / 16;   // 3142
  int wave = blockIdx.x * (blockDim.x >> 5) + (threadIdx.x >> 5);
  if (wave >= TILES) return;
  int lane = threadIdx.x & 31;
  int mb   = wave * 16;
  int m    = mb + (lane & 15);
  if (m >= V) m = V - 1;             // clamp tail loads, stores guarded below
  int koff = (lane >> 4) * 2;

  f32x8 c = {};
  c = wmma_gemv_accum<true>(Wout + (size_t)m * (2 * H), comb, 2 * H, koff, c);

  if ((lane & 15) == 0) {
    int rbase = mb + (lane >> 4) * 8;
#pragma unroll
    for (int r = 0; r < 8; ++r)
      if (rbase + r < V) logits[rbase + r] = c[r] + bout[rbase + r];
  }
}

// ---------------------------------------------------------------------------
// log-softmax over 50257 logits: partial max -> partial sumexp -> write (in place)
// ---------------------------------------------------------------------------
#define LSM_NB 64

__global__ void lsm_pmax_kernel(const float* __restrict__ logits,
                                float* __restrict__ pmax) {
  int i0 = blockIdx.x * 256 + threadIdx.x;
  float m = -INFINITY;
  for (int i = i0; i < V; i += LSM_NB * 256) m = fmaxf(m, logits[i]);
  m = block_reduce_max256(m);
  if (threadIdx.x == 0) pmax[blockIdx.x] = m;
}

__global__ void lsm_psum_kernel(const float* __restrict__ logits,
                                const float* __restrict__ pmax,
                                float* __restrict__ psum) {
  float mv = (threadIdx.x < LSM_NB) ? pmax[threadIdx.x] : -INFINITY;
  float M  = block_reduce_max256(mv);
  int i0 = blockIdx.x * 256 + threadIdx.x;
  float acc = 0.0f;
  for (int i = i0; i < V; i += LSM_NB * 256) acc += expf(logits[i] - M);
  acc = block_reduce_sum256(acc);
  if (threadIdx.x == 0) psum[blockIdx.x] = acc;
}

__global__ void lsm_write_kernel(const float* __restrict__ pmax,
                                 const float* __restrict__ psum,
                                 float* __restrict__ logp) {
  float mv = (threadIdx.x < LSM_NB) ? pmax[threadIdx.x] : -INFINITY;
  float M  = block_reduce_max256(mv);
  float sv = (threadIdx.x < LSM_NB) ? psum[threadIdx.x] : 0.0f;
  float S  = block_reduce_sum256(sv);
  float lse = M + logf(S);
  int i = blockIdx.x * 256 + threadIdx.x;
  if (i < V) logp[i] = logp[i] - lse;   // logits stored in-place in logp region
}

// ---------------------------------------------------------------------------
// Launch
// ---------------------------------------------------------------------------
extern "C" void kernel_launch(void* const* d_in, const int* in_sizes, int n_in,
                              void* d_out, int out_size, void* d_ws, size_t ws_size,
                              hipStream_t stream) {
  const int*   word  = (const int*)  d_in[0];
  const float* h0    = (const float*)d_in[1];
  const float* c0    = (const float*)d_in[2];
  const float* enc   = (const float*)d_in[3];
  const float* emb   = (const float*)d_in[4];
  const float* w_ih  = (const float*)d_in[5];
  const float* w_hh  = (const float*)d_in[6];
  const float* b_ih  = (const float*)d_in[7];
  const float* b_hh  = (const float*)d_in[8];
  const float* Wout  = (const float*)d_in[9];
  const float* bout  = (const float*)d_in[10];

  float* out      = (float*)d_out;
  float* logp     = out;                 // [V]   (logits computed in-place here)
  float* out_h    = out + V;             // [H]
  float* out_c    = out + V + H;         // [H]
  float* out_attn = out + V + 2 * H;     // [SQ]

  float* ws       = (float*)d_ws;
  float* ws_gates = ws;                  // 4096
  float* ws_score = ws + 4096;           // 8192
  float* ws_comb  = ws + 12288;          // 2048
  float* ws_pmax  = ws + 14336;          // 64
  float* ws_psum  = ws + 14400;          // 64
  float* ws_patt  = ws + 14464;          // 32 * 1024

  // 1. gates GEMV: 256 waves
  gates_kernel<<<32, 256, 0, stream>>>(word, emb, h0, w_ih, w_hh, b_ih, b_hh, ws_gates);
  // 2. LSTM pointwise
  lstm_kernel<<<4, 256, 0, stream>>>(ws_gates, c0, out_h, out_c, ws_comb);
  // 3. scores GEMV: 512 waves
  scores_kernel<<<64, 256, 0, stream>>>(enc, ws_comb, ws_score);
  // 4. softmax -> attn (written straight to d_out)
  softmax_kernel<<<1, 256, 0, stream>>>(ws_score, out_attn);
  // 5. attn_out = attn^T @ enc
  attnout_partial_kernel<<<128, 256, 0, stream>>>(out_attn, enc, ws_patt);
  attnout_reduce_kernel<<<4, 256, 0, stream>>>(ws_patt, ws_comb);
  // 6. logits GEMV: 3142 waves -> 393 blocks
  logits_kernel<<<393, 256, 0, stream>>>(Wout, ws_comb, bout, logp);
  // 7. log-softmax (in place on logp region)
  lsm_pmax_kernel<<<LSM_NB, 256, 0, stream>>>(logp, ws_pmax);
  lsm_psum_kernel<<<LSM_NB, 256, 0, stream>>>(logp, ws_pmax, ws_psum);
  lsm_write_kernel<<<(V + 255) / 256, 256, 0, stream>>>(ws_pmax, ws_psum, logp);
}